// GIN_74053826118108
// MI455X (gfx1250) — compile-verified
//
#include <hip/hip_runtime.h>
#include <hip/hip_bf16.h>

// ---------------------------------------------------------------------------
// GIN forward for MI455X (gfx1250, wave32).
// fp32 WMMA (V_WMMA_F32_16X16X4_F32) for all dense GEMMs; atomics for graph
// scatter/pool (L2-resident working set ~40 MB << 192 MB L2).
// Round 3: W staged into LDS in k-pair-interleaved, bank-padded layout so
// every B fragment is one aligned ds_load_b64 (no register shuffles).
// ---------------------------------------------------------------------------

typedef float v2f __attribute__((ext_vector_type(2)));
typedef float v8f __attribute__((ext_vector_type(8)));

#define G_GRAPHS 256
#define HDIM     64
#define NCLS     10

// k-pair stride in floats: 160 ≡ 32 (mod 64 banks) -> the two lane-halves
// (reading pair p and p+1) use disjoint bank sets for their b64 loads.
#define WP_STRIDE 160

enum { FLAG_RELU_A = 1, FLAG_RELU_OUT = 2, FLAG_ADD_A2 = 4, FLAG_ACCUM = 8 };

// ---------------------------------------------------------------------------
// Fused WMMA GEMM: Out[M x 64] = epilogue( A[M x K] @ W[K x 64] + bias )
//   FLAG_RELU_A  : relu applied to A elements feeding the WMMA
//   FLAG_RELU_OUT: relu(res + bias)
//   FLAG_ADD_A2  : add A2[row*64+col] after activation
//   FLAG_ACCUM   : Out += val   (else Out = val)
// 4 waves / block, each wave computes one 16x64 tile (4 accumulators).
// M is always a multiple of 16 here, so waves are entirely valid or invalid
// and EXEC stays all-ones for the WMMA ops.
// ---------------------------------------------------------------------------
template <int FLAGS, int K>
__global__ void __launch_bounds__(128) k_wmma_gemm64(
    const float* __restrict__ A, int M,
    const float* __restrict__ W, const float* __restrict__ bias,
    const float* __restrict__ A2, float* __restrict__ Out)
{
    // sWp[p][col*2 + j] = W[2p + j][col]  (j = 0,1), padded stride
    __shared__ float sWp[(K / 2) * WP_STRIDE];
    __shared__ float sB[64];
    const int tid = threadIdx.x;
#pragma unroll
    for (int p = 0; p < (K / 2) * 64; p += 128) {
        const int p2  = (p + tid) >> 6;        // k-pair index
        const int col = (p + tid) & 63;
        v2f w;
        w.x = W[(p2 * 2)     * 64 + col];
        w.y = W[(p2 * 2 + 1) * 64 + col];
        *(v2f*)&sWp[p2 * WP_STRIDE + col * 2] = w;
    }
    if (tid < 64) sB[tid] = bias ? bias[tid] : 0.0f;
    __syncthreads();

    const int wave  = tid >> 5;          // wave32
    const int lane  = tid & 31;
    const int mBase = (blockIdx.x * 4 + wave) * 16;
    if (mBase >= M) return;              // whole-wave uniform exit (after barrier)

    const int r    = lane & 15;          // M row within tile (A) / N col (B,C,D)
    const int kSel = (lane >> 4) << 1;   // lanes 0-15 -> K 0,1 ; lanes 16-31 -> K 2,3
    const float* Arow = A + (size_t)(mBase + r) * K;

    v8f acc[4];
#pragma unroll
    for (int nt = 0; nt < 4; ++nt) acc[nt] = 0.0f;

#pragma unroll
    for (int kt = 0; kt < K; kt += 4) {
        v2f a;
        a.x = Arow[kt + kSel];
        a.y = Arow[kt + kSel + 1];
        if constexpr (FLAGS & FLAG_RELU_A) {
            a.x = fmaxf(a.x, 0.0f);
            a.y = fmaxf(a.y, 0.0f);
        }
        const int pRow = ((kt + kSel) >> 1) * WP_STRIDE;
#pragma unroll
        for (int nt = 0; nt < 4; ++nt) {
            const int col = nt * 16 + r;
            // single aligned ds_load_b64 -> even VGPR pair
            const v2f b = *(const v2f*)&sWp[pRow + col * 2];
            // D = A(16x4) * B(4x16) + C, fp32
            acc[nt] = __builtin_amdgcn_wmma_f32_16x16x4_f32(
                false, a, false, b, (short)0, acc[nt], false, false);
        }
    }

    // C/D layout: VGPR v -> M = v + 8*(lane>=16), N = lane & 15
    const int rowOff = (lane >> 4) << 3;
#pragma unroll
    for (int nt = 0; nt < 4; ++nt) {
        const int col = nt * 16 + r;
#pragma unroll
        for (int v = 0; v < 8; ++v) {
            const int row = mBase + rowOff + v;
            float val = acc[nt][v] + sB[col];
            if constexpr (FLAGS & FLAG_RELU_OUT) val = fmaxf(val, 0.0f);
            if constexpr (FLAGS & FLAG_ADD_A2)   val += A2[(size_t)row * 64 + col];
            float* o = Out + (size_t)row * 64 + col;
            if constexpr (FLAGS & FLAG_ACCUM) *o += val; else *o = val;
        }
    }
}

// ---------------------------------------------------------------------------
// BatchNorm statistics: per-column sum / sumsq -> sums[0:64], sums[64:128]
// ---------------------------------------------------------------------------
__global__ void __launch_bounds__(256) k_bn_stats(
    const float* __restrict__ h, int n, float* __restrict__ sums)
{
    const int col = threadIdx.x & 63;
    const int sub = threadIdx.x >> 6;    // 0..3
    float s = 0.0f, s2 = 0.0f;
    for (int row = blockIdx.x * 4 + sub; row < n; row += gridDim.x * 4) {
        const float v = h[(size_t)row * 64 + col];
        s += v; s2 += v * v;
    }
    __shared__ float ls[256], ls2[256];
    ls[threadIdx.x] = s; ls2[threadIdx.x] = s2;
    __syncthreads();
    if (threadIdx.x < 64) {
        const int c = threadIdx.x;
        s  = ls[c]  + ls[c + 64]  + ls[c + 128]  + ls[c + 192];
        s2 = ls2[c] + ls2[c + 64] + ls2[c + 128] + ls2[c + 192];
        atomicAdd(&sums[c], s);
        atomicAdd(&sums[64 + c], s2);
    }
}

__global__ void k_bn_finalize(const float* __restrict__ sums,
                              const float* __restrict__ g,
                              const float* __restrict__ b,
                              int n, float* __restrict__ scsh)
{
    const int c = threadIdx.x;           // 64 threads
    const float inv_n = 1.0f / (float)n;
    const float mean = sums[c] * inv_n;
    const float var  = sums[64 + c] * inv_n - mean * mean;
    const float sc   = g[c] / sqrtf(var + 1e-5f);
    scsh[c]      = sc;
    scsh[64 + c] = b[c] - mean * sc;
}

__global__ void __launch_bounds__(256) k_bn_apply(
    const float* __restrict__ h, const float* __restrict__ scsh,
    float* __restrict__ o, int total /* multiple of 4 */)
{
    const int idx = (blockIdx.x * 256 + threadIdx.x) * 4;
    if (idx >= total) return;
    float4 v = *(const float4*)&h[idx];
    const int c = idx & 63;
    v.x = v.x * scsh[c]     + scsh[64 + c];
    v.y = v.y * scsh[c + 1] + scsh[64 + c + 1];
    v.z = v.z * scsh[c + 2] + scsh[64 + c + 2];
    v.w = v.w * scsh[c + 3] + scsh[64 + c + 3];
    *(float4*)&o[idx] = v;
}

// ---------------------------------------------------------------------------
// Edge scatter-add: agg[dst] += h[src]   (64 threads per edge)
// ---------------------------------------------------------------------------
__global__ void __launch_bounds__(256) k_scatter(
    const float* __restrict__ h, const int* __restrict__ src,
    const int* __restrict__ dst, int nE, float* __restrict__ agg)
{
    const int e = blockIdx.x * 4 + (threadIdx.x >> 6);
    const int f = threadIdx.x & 63;
    if (e < nE) {
        const int s = src[e], d = dst[e];
        atomicAdd(&agg[(size_t)d * 64 + f], h[(size_t)s * 64 + f]);
    }
}

// out = a + b (elementwise, float4); out may alias b
__global__ void __launch_bounds__(256) k_add(
    const float* __restrict__ a, const float* __restrict__ b,
    float* __restrict__ o, int total)
{
    const int idx = (blockIdx.x * 256 + threadIdx.x) * 4;
    if (idx >= total) return;
    const float4 va = *(const float4*)&a[idx];
    const float4 vb = *(const float4*)&b[idx];
    float4 v; v.x = va.x + vb.x; v.y = va.y + vb.y; v.z = va.z + vb.z; v.w = va.w + vb.w;
    *(float4*)&o[idx] = v;
}

__global__ void k_count(const int* __restrict__ batch, int n, float* __restrict__ cnt)
{
    const int i = blockIdx.x * 256 + threadIdx.x;
    if (i < n) atomicAdd(&cnt[batch[i]], 1.0f);
}

// pool[batch[row]*F + f] += h[row*F + f]
__global__ void __launch_bounds__(256) k_pool(
    const float* __restrict__ h, const int* __restrict__ batch,
    int n, int F, float* __restrict__ pool)
{
    const int rpb = 256 / F;
    const int row = blockIdx.x * rpb + threadIdx.x / F;
    const int f   = threadIdx.x & (F - 1);
    if (row < n) {
        const int g = batch[row];
        atomicAdd(&pool[(size_t)g * F + f], h[(size_t)row * F + f]);
    }
}

__global__ void k_pool_div(float* __restrict__ pool, const float* __restrict__ cnt,
                           int F, int total)
{
    const int i = blockIdx.x * 256 + threadIdx.x;
    if (i < total) pool[i] /= fmaxf(cnt[i / F], 1.0f);
}

// ---------------------------------------------------------------------------
// Head: o = relu(out)/4; a = relu(o@afW+afB)+o; l = a@fW+fB; log_softmax
// One block (64 threads) per graph.
// ---------------------------------------------------------------------------
__global__ void __launch_bounds__(64) k_head(
    const float* __restrict__ outG, const float* __restrict__ afW,
    const float* __restrict__ afB, const float* __restrict__ fW,
    const float* __restrict__ fB, float* __restrict__ res)
{
    __shared__ float so[64], sa[64], sl[NCLS];
    const int g = blockIdx.x, t = threadIdx.x;
    const float o = fmaxf(outG[(size_t)g * 64 + t], 0.0f) * 0.25f;
    so[t] = o;
    __syncthreads();
    float s = afB[t];
    for (int k = 0; k < 64; ++k) s += so[k] * afW[k * 64 + t];
    sa[t] = fmaxf(s, 0.0f) + o;
    __syncthreads();
    if (t < NCLS) {
        float l = fB[t];
        for (int k = 0; k < 64; ++k) l += sa[k] * fW[k * NCLS + t];
        sl[t] = l;
    }
    __syncthreads();
    if (t == 0) {
        float m = sl[0];
        for (int c = 1; c < NCLS; ++c) m = fmaxf(m, sl[c]);
        float se = 0.0f;
        for (int c = 0; c < NCLS; ++c) se += expf(sl[c] - m);
        const float lse = m + logf(se);
        for (int c = 0; c < NCLS; ++c) res[(size_t)g * NCLS + c] = sl[c] - lse;
    }
}

// ---------------------------------------------------------------------------
// Host orchestration
// ---------------------------------------------------------------------------
extern "C" void kernel_launch(void* const* d_in, const int* in_sizes, int n_in,
                              void* d_out, int out_size, void* d_ws, size_t ws_size,
                              hipStream_t stream)
{
    (void)n_in; (void)out_size; (void)ws_size;
    // input order: x, noprop{xg_w,xg_b,lin_w}, init_w, init_b,
    //              layers[0..3]{l1_w,l1_b,l2_w,l2_b,xg_w,xg_b,lin_w,bn_g,bn_b},
    //              after_w, after_b, final_w, final_b, edge_index, batch, num_graphs
    const float* x       = (const float*)d_in[0];
    const float* np_xgw  = (const float*)d_in[1];
    const float* np_xgb  = (const float*)d_in[2];
    const float* np_linw = (const float*)d_in[3];
    const float* init_w  = (const float*)d_in[4];
    const float* init_b  = (const float*)d_in[5];
    const float* after_w = (const float*)d_in[42];
    const float* after_b = (const float*)d_in[43];
    const float* final_w = (const float*)d_in[44];
    const float* final_b = (const float*)d_in[45];
    const int*   edge    = (const int*)d_in[46];
    const int*   batch   = (const int*)d_in[47];

    const int N = in_sizes[0] / 32;       // 50000 (multiple of 16)
    const int E = in_sizes[46] / 2;       // 800000
    const int* src = edge;
    const int* dst = edge + E;

    const size_t N64 = (size_t)N * 64;
    float* A      = (float*)d_ws;         // h
    float* B      = A + N64;              // hb / u
    float* C      = B + N64;              // agg / z
    float* pool   = C + N64;              // G*64
    float* gbuf   = pool + G_GRAPHS * 64; // G*64
    float* outG   = gbuf + G_GRAPHS * 64; // G*64
    float* cnt    = outG + G_GRAPHS * 64; // G
    float* bnsums = cnt + G_GRAPHS;       // 128
    float* scsh   = bnsums + 128;         // 128

    auto blocks = [](int M) { return dim3((M + 63) / 64); };

    // graph sizes
    hipMemsetAsync(cnt, 0, G_GRAPHS * sizeof(float), stream);
    k_count<<<dim3((N + 255) / 256), dim3(256), 0, stream>>>(batch, N, cnt);

    // no-prop readout on raw x:  out = g + relu(g)@lin_w, g = mean(x)@xg_w + xg_b
    hipMemsetAsync(pool, 0, G_GRAPHS * 32 * sizeof(float), stream);
    k_pool<<<dim3((N + 7) / 8), dim3(256), 0, stream>>>(x, batch, N, 32, pool);
    k_pool_div<<<dim3((G_GRAPHS * 32 + 255) / 256), dim3(256), 0, stream>>>(
        pool, cnt, 32, G_GRAPHS * 32);
    k_wmma_gemm64<0, 32><<<blocks(G_GRAPHS), dim3(128), 0, stream>>>(
        pool, G_GRAPHS, np_xgw, np_xgb, nullptr, gbuf);
    k_wmma_gemm64<FLAG_RELU_A | FLAG_ADD_A2, 64><<<blocks(G_GRAPHS), dim3(128), 0, stream>>>(
        gbuf, G_GRAPHS, np_linw, nullptr, gbuf, outG);   // outG = g + relu(g)@lin_w

    // h = x @ init_w + init_b
    k_wmma_gemm64<0, 32><<<blocks(N), dim3(128), 0, stream>>>(
        x, N, init_w, init_b, nullptr, A);

    for (int i = 0; i < 4; ++i) {
        const float* l1w = (const float*)d_in[6 + 9 * i + 0];
        const float* l1b = (const float*)d_in[6 + 9 * i + 1];
        const float* l2w = (const float*)d_in[6 + 9 * i + 2];
        const float* l2b = (const float*)d_in[6 + 9 * i + 3];
        const float* xgw = (const float*)d_in[6 + 9 * i + 4];
        const float* xgb = (const float*)d_in[6 + 9 * i + 5];
        const float* lnw = (const float*)d_in[6 + 9 * i + 6];
        const float* bng = (const float*)d_in[6 + 9 * i + 7];
        const float* bnb = (const float*)d_in[6 + 9 * i + 8];

        const float* hb = A;
        if (i > 0) {  // training-mode BatchNorm
            hipMemsetAsync(bnsums, 0, 128 * sizeof(float), stream);
            k_bn_stats<<<dim3(256), dim3(256), 0, stream>>>(A, N, bnsums);
            k_bn_finalize<<<dim3(1), dim3(64), 0, stream>>>(bnsums, bng, bnb, N, scsh);
            k_bn_apply<<<dim3((int)(N64 / 4 + 255) / 256), dim3(256), 0, stream>>>(
                A, scsh, B, (int)N64);
            hb = B;
        }

        // agg = segment_sum(hb[src], dst)
        hipMemsetAsync(C, 0, N64 * sizeof(float), stream);
        k_scatter<<<dim3((E + 3) / 4), dim3(256), 0, stream>>>(hb, src, dst, E, C);
        // z = hb + agg   (in place into C)
        k_add<<<dim3((int)(N64 / 4 + 255) / 256), dim3(256), 0, stream>>>(
            hb, C, C, (int)N64);
        // u = z + relu(z @ l1_w + l1_b)
        k_wmma_gemm64<FLAG_RELU_OUT | FLAG_ADD_A2, 64><<<blocks(N), dim3(128), 0, stream>>>(
            C, N, l1w, l1b, C, B);
        // h = u @ l2_w + l2_b
        k_wmma_gemm64<0, 64><<<blocks(N), dim3(128), 0, stream>>>(
            B, N, l2w, l2b, nullptr, A);

        // out += g + relu(g)@lin_w, g = mean(h)@xg_w + xg_b
        hipMemsetAsync(pool, 0, G_GRAPHS * 64 * sizeof(float), stream);
        k_pool<<<dim3((N + 3) / 4), dim3(256), 0, stream>>>(A, batch, N, 64, pool);
        k_pool_div<<<dim3((G_GRAPHS * 64 + 255) / 256), dim3(256), 0, stream>>>(
            pool, cnt, 64, G_GRAPHS * 64);
        k_wmma_gemm64<0, 64><<<blocks(G_GRAPHS), dim3(128), 0, stream>>>(
            pool, G_GRAPHS, xgw, xgb, nullptr, gbuf);
        k_wmma_gemm64<FLAG_RELU_A | FLAG_ADD_A2 | FLAG_ACCUM, 64>
            <<<blocks(G_GRAPHS), dim3(128), 0, stream>>>(
            gbuf, G_GRAPHS, lnw, nullptr, gbuf, outG);
    }

    // head: relu/L -> after-linear residual -> final linear -> log_softmax
    k_head<<<dim3(G_GRAPHS), dim3(64), 0, stream>>>(
        outG, after_w, after_b, final_w, final_b, (float*)d_out);
}